// AleatoricCrossEntropyLoss_35742717838029
// MI455X (gfx1250) — compile-verified
//
#include <hip/hip_runtime.h>
#include <hip/hip_bf16.h>
#include <stdint.h>

// ---------------------------------------------------------------------------
// Aleatoric CE loss, MC T=100, B=4096, C=8192 (fp32).
// Strategy: 1 block per row; row staged to LDS once via async global->LDS
// (ASYNCcnt); 101 softmax passes (1 undistorted + 100 MC) out of LDS.
// Cross-lane sum reduction uses V_WMMA_F32_16X16X4_F32 (B = ones).
// Per-trial sums accumulated with global fp32 atomics into d_ws[0..101].
// ---------------------------------------------------------------------------

#define B_ROWS 4096
#define C_COLS 8192
#define T_MC   100
#define NTHR   256
#define NWAVES (NTHR / 32)
#define SLOTS  (C_COLS / NTHR)   // 32 elements per thread

typedef __attribute__((ext_vector_type(2))) float v2f;
typedef __attribute__((ext_vector_type(8))) float v8f;

// 32-lane sum via WMMA: A[16x4] = lane partials (k-odd entries 0), B = ones.
// D[m][n] = s[m] + s[m+16]; summing the 8 D VGPRs per lane gives half-wave
// sums; one xor-16 shuffle finishes. EXEC must be all ones (uniform call site).
__device__ __forceinline__ float wave_sum_wmma(float x) {
  v2f a; a[0] = x;    a[1] = 0.0f;
  v2f b; b[0] = 1.0f; b[1] = 1.0f;
  v8f c = {};
  v8f d = __builtin_amdgcn_wmma_f32_16x16x4_f32(
      /*neg_a=*/false, a, /*neg_b=*/false, b,
      /*c_mod=*/(short)0, c, /*reuse_a=*/false, /*reuse_b=*/false);
  float s = ((d[0] + d[1]) + (d[2] + d[3])) + ((d[4] + d[5]) + (d[6] + d[7]));
  s += __shfl_xor(s, 16, 32);
  return s;
}

__device__ __forceinline__ float wave_max(float x) {
#pragma unroll
  for (int off = 16; off > 0; off >>= 1)
    x = fmaxf(x, __shfl_xor(x, off, 32));
  return x;
}

__device__ __forceinline__ void philox4x32(uint32_t c0, uint32_t c1,
                                           uint32_t c2, uint32_t c3,
                                           uint32_t k0, uint32_t k1,
                                           uint32_t out[4]) {
#pragma unroll
  for (int i = 0; i < 10; ++i) {
    uint32_t lo0 = 0xD2511F53u * c0;
    uint32_t hi0 = __umulhi(0xD2511F53u, c0);
    uint32_t lo1 = 0xCD9E8D57u * c2;
    uint32_t hi1 = __umulhi(0xCD9E8D57u, c2);
    uint32_t n0 = hi1 ^ c1 ^ k0;
    uint32_t n2 = hi0 ^ c3 ^ k1;
    c0 = n0; c1 = lo1; c2 = n2; c3 = lo0;
    k0 += 0x9E3779B9u; k1 += 0xBB67AE85u;
  }
  out[0] = c0; out[1] = c1; out[2] = c2; out[3] = c3;
}

__device__ __forceinline__ void box_muller(uint32_t a, uint32_t b,
                                           float& n0, float& n1) {
  // u1 in (0,1], u2 in [0,1)
  float u1 = (float)((a >> 8) + 1u) * (1.0f / 16777216.0f);
  float u2 = (float)(b >> 8) * (1.0f / 16777216.0f);
  float rad = __fsqrt_rn(-2.0f * __logf(u1));
  float ang = 6.28318530718f * u2;
  n0 = rad * __cosf(ang);
  n1 = rad * __sinf(ang);
}

__global__ void acel_zero(float* acc) {
  int i = threadIdx.x;
  if (i < T_MC + 2) acc[i] = 0.0f;
}

__global__ __launch_bounds__(NTHR)
void acel_main(const float* __restrict__ logit_var,
               const float* __restrict__ logit,
               const int*   __restrict__ labels,
               float*       __restrict__ acc) {
  __shared__ float row[C_COLS];       // 32 KB: one logit row, reused 101x
  __shared__ float redA[NWAVES];
  __shared__ float redB[NWAVES];

  const int tid  = threadIdx.x;
  const int wid  = tid >> 5;
  const int lane = tid & 31;
  const int r    = blockIdx.x;

  // ---- async stage: global -> LDS, 8 x b128 per lane (ASYNCcnt) ----
  {
    const float* g  = logit + (size_t)r * C_COLS;
    uint64_t gbase  = (uint64_t)(uintptr_t)g;
    unsigned lbase  = (unsigned)(uintptr_t)(&row[0]);  // low 32b = LDS addr
#pragma unroll
    for (int k = 0; k < 8; ++k) {
      unsigned off = (unsigned)(tid * 16 + k * (NTHR * 16));
      unsigned la  = lbase + off;
      uint64_t ga  = gbase + off;
      asm volatile("global_load_async_to_lds_b128 %0, %1, off"
                   :: "v"(la), "v"(ga) : "memory");
    }
    asm volatile("s_wait_asynccnt 0" ::: "memory");
  }
  __syncthreads();

  const float lv   = logit_var[r];
  const float stdv = __fsqrt_rn(lv);
  const int   lab  = labels[r];
  const bool  isLab   = (tid == (lab & (NTHR - 1)));
  const int   labSlot = lab >> 8;     // col = (lab&255) + 256*slot

  // ---- undistorted pass (eps = 0) ----
  {
    float m = -3.402823466e38f;
#pragma unroll
    for (int s = 0; s < SLOTS; ++s) m = fmaxf(m, row[tid + NTHR * s]);
    m = wave_max(m);
    if (lane == 0) redA[wid] = m;
    __syncthreads();
    float rowmax = redA[0];
#pragma unroll
    for (int w = 1; w < NWAVES; ++w) rowmax = fmaxf(rowmax, redA[w]);

    float s0 = 0.0f;
#pragma unroll
    for (int s = 0; s < SLOTS; ++s) s0 += __expf(row[tid + NTHR * s] - rowmax);
    s0 = wave_sum_wmma(s0);
    if (lane == 0) redB[wid] = s0;
    __syncthreads();
    float sum = 0.0f;
#pragma unroll
    for (int w = 0; w < NWAVES; ++w) sum += redB[w];

    if (isLab)
      atomicAdd(&acc[T_MC], -(row[lab] - rowmax - __logf(sum)));
    if (tid == 0)
      atomicAdd(&acc[T_MC + 1], __expf(lv) - 1.0f);
  }

  // ---- 100 Monte-Carlo passes out of LDS ----
  for (int t = 0; t < T_MC; ++t) {
    float v[SLOTS];
    float dlab = 0.0f;
    float m = -3.402823466e38f;
#pragma unroll
    for (int k = 0; k < SLOTS / 4; ++k) {
      uint32_t rnd[4];
      philox4x32((uint32_t)r, (uint32_t)t, (uint32_t)tid, (uint32_t)k,
                 0xCAFEF00Du, 0xDEADBEEFu, rnd);
      float n01[4];
      box_muller(rnd[0], rnd[1], n01[0], n01[1]);
      box_muller(rnd[2], rnd[3], n01[2], n01[3]);
#pragma unroll
      for (int j = 0; j < 4; ++j) {
        const int s = 4 * k + j;          // unroll-constant index
        float d = row[tid + NTHR * s] + n01[j] * stdv;
        v[s] = d;
        if (s == labSlot) dlab = d;       // capture label value, no dyn index
        m = fmaxf(m, d);
      }
    }
    m = wave_max(m);
    if (lane == 0) redA[wid] = m;
    __syncthreads();
    float rowmax = redA[0];
#pragma unroll
    for (int w = 1; w < NWAVES; ++w) rowmax = fmaxf(rowmax, redA[w]);

    float s0 = 0.0f;
#pragma unroll
    for (int s = 0; s < SLOTS; ++s) s0 += __expf(v[s] - rowmax);
    s0 = wave_sum_wmma(s0);
    if (lane == 0) redB[wid] = s0;
    __syncthreads();
    float sum = 0.0f;
#pragma unroll
    for (int w = 0; w < NWAVES; ++w) sum += redB[w];

    if (isLab)
      atomicAdd(&acc[t], -(dlab - rowmax - __logf(sum)));
  }
}

__global__ void acel_finish(const float* __restrict__ acc,
                            float* __restrict__ out) {
  __shared__ float sg[128];
  __shared__ float sv[128];
  const int tid = threadIdx.x;
  const float invB = 1.0f / (float)B_ROWS;
  const float undist = acc[T_MC] * invB;

  float g = 0.0f, vl = 0.0f;
  if (tid < T_MC) {
    float dl = acc[tid] * invB;
    g = dl;
    float diff = undist - dl;
    float elu = diff > 0.0f ? diff : (__expf(diff) - 1.0f);
    vl = -elu;
  }
  sg[tid] = g; sv[tid] = vl;
  __syncthreads();
  for (int off = 64; off > 0; off >>= 1) {
    if (tid < off) { sg[tid] += sg[tid + off]; sv[tid] += sv[tid + off]; }
    __syncthreads();
  }
  if (tid == 0) {
    out[0] = sg[0] * (1.0f / (float)T_MC);   // gce_loss
    out[1] = sv[0] * (1.0f / (float)T_MC);   // variance_loss
    out[2] = undist;                          // undistorted_loss
    out[3] = acc[T_MC + 1] * invB;            // variance_depressor
  }
}

extern "C" void kernel_launch(void* const* d_in, const int* in_sizes, int n_in,
                              void* d_out, int out_size, void* d_ws, size_t ws_size,
                              hipStream_t stream) {
  (void)in_sizes; (void)n_in; (void)out_size; (void)ws_size;
  const float* logit_var = (const float*)d_in[0];
  const float* logit     = (const float*)d_in[1];
  const int*   labels    = (const int*)d_in[2];
  float* out = (float*)d_out;
  float* acc = (float*)d_ws;   // needs (T_MC+2) floats

  acel_zero  <<<1, 128, 0, stream>>>(acc);
  acel_main  <<<B_ROWS, NTHR, 0, stream>>>(logit_var, logit, labels, acc);
  acel_finish<<<1, 128, 0, stream>>>(acc, out);
}